// InfoNCECauchy_47528108098304
// MI455X (gfx1250) — compile-verified
//
#include <hip/hip_runtime.h>
#include <math.h>

typedef __attribute__((ext_vector_type(2))) float v2f;
typedef __attribute__((ext_vector_type(8))) float v8f;

#define NHALF 8192
#define DIM   128
#define BLK   128
#define NBLK  64          // 8192/128
#define TRI   2080        // 64*65/2
#define LSTR  132         // LDS row stride (floats): 16B-aligned rows, conflict-free b64 frag reads

__global__ __launch_bounds__(256) void zero_ws_kernel(float* p, int n) {
  int i = blockIdx.x * 256 + threadIdx.x;
  if (i < n) p[i] = 0.f;
}

__global__ __launch_bounds__(256) void row_norms_kernel(const float* __restrict__ X,
                                                        float* __restrict__ sq) {
  int i = blockIdx.x * 256 + threadIdx.x;   // grid covers 16384 rows exactly
  const float4* p = (const float4*)(X + (size_t)i * DIM);
  float s = 0.f;
#pragma unroll
  for (int k = 0; k < DIM / 4; ++k) {
    float4 v = p[k];
    s += v.x * v.x + v.y * v.y + v.z * v.z + v.w * v.w;
  }
  sq[i] = s;
}

__global__ __launch_bounds__(256) void cauchy_main_kernel(
    const float* __restrict__ X, const float* __restrict__ sq,
    float* __restrict__ rs_aa, float* __restrict__ rs_bb,
    float* __restrict__ rs_ab, float* __restrict__ cs_ab,
    float* __restrict__ ta) {
  __shared__ float As[BLK * LSTR];
  __shared__ float Bs[BLK * LSTR];

  const int tid  = threadIdx.x;
  const int lane = tid & 31;
  const int wave = tid >> 5;

  // ---- decode block -> (type, bm, bn); aa/bb only upper triangle ----
  int b = blockIdx.x;
  int type, bm, bn;
  if (b < 2 * TRI) {
    type = (b < TRI) ? 0 : 1;
    int tri = (type == 0) ? b : (b - TRI);
    bm = 0;
    while (tri >= (NBLK - bm)) { tri -= (NBLK - bm); ++bm; }
    bn = bm + tri;
  } else {
    type = 2;
    int q = b - 2 * TRI;
    bm = q >> 6;
    bn = q & 63;
  }
  const int rowBase = bm * BLK + (type == 1 ? NHALF : 0);
  const int colBase = bn * BLK + (type >= 1 ? NHALF : 0);

  // ---- cooperative load of 128x128 A and B blocks into LDS ----
  for (int i = tid; i < BLK * (DIM / 4); i += 256) {
    int row = i >> 5;
    int c4  = (i & 31) << 2;
    float4 va = *(const float4*)(X + (size_t)(rowBase + row) * DIM + c4);
    float4 vb = *(const float4*)(X + (size_t)(colBase + row) * DIM + c4);
    *(float4*)(&As[row * LSTR + c4]) = va;
    *(float4*)(&Bs[row * LSTR + c4]) = vb;
  }
  __syncthreads();

  // ---- per-wave 32x64 tile: 2x4 accumulators ----
  const int mOff = (wave >> 1) * 32;
  const int nOff = (wave & 1) * 64;
  const int lrow = lane & 15;
  const int kHi  = (lane >> 4) << 1;   // lanes 16-31 hold K+2,K+3

  v8f acc[2][4];
#pragma unroll
  for (int i = 0; i < 2; ++i)
#pragma unroll
    for (int j = 0; j < 4; ++j)
      acc[i][j] = (v8f){0.f, 0.f, 0.f, 0.f, 0.f, 0.f, 0.f, 0.f};

  for (int k0 = 0; k0 < DIM; k0 += 4) {
    const int kk = k0 + kHi;
    v2f a0 = *(const v2f*)(&As[(mOff +  0 + lrow) * LSTR + kk]);
    v2f a1 = *(const v2f*)(&As[(mOff + 16 + lrow) * LSTR + kk]);
    v2f b0 = *(const v2f*)(&Bs[(nOff +  0 + lrow) * LSTR + kk]);
    v2f b1 = *(const v2f*)(&Bs[(nOff + 16 + lrow) * LSTR + kk]);
    v2f b2 = *(const v2f*)(&Bs[(nOff + 32 + lrow) * LSTR + kk]);
    v2f b3 = *(const v2f*)(&Bs[(nOff + 48 + lrow) * LSTR + kk]);
    acc[0][0] = __builtin_amdgcn_wmma_f32_16x16x4_f32(false, a0, false, b0, (short)0, acc[0][0], false, false);
    acc[0][1] = __builtin_amdgcn_wmma_f32_16x16x4_f32(false, a0, false, b1, (short)0, acc[0][1], false, false);
    acc[0][2] = __builtin_amdgcn_wmma_f32_16x16x4_f32(false, a0, false, b2, (short)0, acc[0][2], false, false);
    acc[0][3] = __builtin_amdgcn_wmma_f32_16x16x4_f32(false, a0, false, b3, (short)0, acc[0][3], false, false);
    acc[1][0] = __builtin_amdgcn_wmma_f32_16x16x4_f32(false, a1, false, b0, (short)0, acc[1][0], false, false);
    acc[1][1] = __builtin_amdgcn_wmma_f32_16x16x4_f32(false, a1, false, b1, (short)0, acc[1][1], false, false);
    acc[1][2] = __builtin_amdgcn_wmma_f32_16x16x4_f32(false, a1, false, b2, (short)0, acc[1][2], false, false);
    acc[1][3] = __builtin_amdgcn_wmma_f32_16x16x4_f32(false, a1, false, b3, (short)0, acc[1][3], false, false);
  }

  // ---- reuse LDS for block-level row/col accumulators ----
  __syncthreads();
  float* rAcc = As;          // 128 row sums (local M)
  float* cAcc = As + BLK;    // 128 col sums (local N)
  As[tid] = 0.f;             // tid in [0,256) covers both
  __syncthreads();

#pragma unroll
  for (int mt = 0; mt < 2; ++mt) {
    const int mHalf = mOff + 16 * mt + ((lane >> 4) << 3);
#pragma unroll
    for (int nt = 0; nt < 4; ++nt) {
      const int nLoc = nOff + 16 * nt + lrow;
      const int gN   = colBase + nLoc;
      const float sqn = sq[gN];
      float colpart = 0.f;
      v8f cv = acc[mt][nt];
#pragma unroll
      for (int r = 0; r < 8; ++r) {
        const int mLoc = mHalf + r;
        const int gM   = rowBase + mLoc;
        float d2 = sq[gM] + sqn - 2.f * cv[r];
        d2 = fmaxf(d2, 0.f);
        float s = 1.f / (d2 + 1.f);                 // TEMPERATURE = 1
        if (type < 2 && gM == gN) s = 0.f;          // zero aa/bb diagonal
        if (type == 2 && (gM + NHALF) == gN)        // ab diagonal -> alignment term
          atomicAdd(ta, logf(s));
        colpart += s;
        float rv = s;                                // reduce over 16 lanes (N within tile)
        rv += __shfl_xor(rv, 1, 16);
        rv += __shfl_xor(rv, 2, 16);
        rv += __shfl_xor(rv, 4, 16);
        rv += __shfl_xor(rv, 8, 16);
        if (lrow == 0) atomicAdd(&rAcc[mLoc], rv);
      }
      atomicAdd(&cAcc[nLoc], colpart);
    }
  }
  __syncthreads();

  // ---- one global atomic per row/col per workgroup ----
  if (tid < BLK) {
    float v = rAcc[tid];
    float* rt = (type == 0) ? rs_aa : (type == 1) ? rs_bb : rs_ab;
    atomicAdd(&rt[bm * BLK + tid], v);
  } else {
    int t2 = tid - BLK;
    float v = cAcc[t2];
    if (type == 2) {
      atomicAdd(&cs_ab[bn * BLK + t2], v);
    } else if (bm != bn) {                   // symmetric: col sums feed row sums
      float* rt = (type == 0) ? rs_aa : rs_bb;
      atomicAdd(&rt[bn * BLK + t2], v);
    }
  }
}

__global__ __launch_bounds__(256) void finalize_kernel(
    const float* __restrict__ rs_aa, const float* __restrict__ rs_bb,
    const float* __restrict__ rs_ab, const float* __restrict__ cs_ab,
    const float* __restrict__ ta, float* __restrict__ out) {
  __shared__ float red[256];
  float acc = 0.f;
  for (int i = threadIdx.x; i < NHALF; i += 256) {
    acc += logf(rs_aa[i] + rs_ab[i]);   // logsumexp_2 terms
    acc += logf(cs_ab[i] + rs_bb[i]);   // logsumexp_1 terms
  }
  red[threadIdx.x] = acc;
  __syncthreads();
  for (int s = 128; s > 0; s >>= 1) {
    if (threadIdx.x < s) red[threadIdx.x] += red[threadIdx.x + s];
    __syncthreads();
  }
  if (threadIdx.x == 0) {
    const float n = (float)NHALF;
    float alignment  = ta[0] / n;          // mean log diag(sim_ab), EXAGGERATION = 1
    float uniformity = red[0] / n;         // logsumexp_1 + logsumexp_2
    out[0] = -(alignment - 0.5f * uniformity);
  }
}

extern "C" void kernel_launch(void* const* d_in, const int* in_sizes, int n_in,
                              void* d_out, int out_size, void* d_ws, size_t ws_size,
                              hipStream_t stream) {
  const float* X = (const float*)d_in[0];   // 16384 x 128 fp32
  float* out = (float*)d_out;
  float* ws  = (float*)d_ws;

  // workspace layout (floats)
  float* sq    = ws;                 // 16384 row norms
  float* rs_aa = ws + 16384;         // 8192
  float* rs_bb = rs_aa + NHALF;      // 8192
  float* rs_ab = rs_bb + NHALF;      // 8192
  float* cs_ab = rs_ab + NHALF;      // 8192
  float* ta    = cs_ab + NHALF;      // 1 (sum of log diag sim_ab)
  const int nzero = 4 * NHALF + 1;   // zero the accumulators (sq fully overwritten)

  zero_ws_kernel<<<(nzero + 255) / 256, 256, 0, stream>>>(rs_aa, nzero);
  row_norms_kernel<<<16384 / 256, 256, 0, stream>>>(X, sq);
  cauchy_main_kernel<<<2 * TRI + NBLK * NBLK, 256, 0, stream>>>(X, sq, rs_aa, rs_bb, rs_ab, cs_ab, ta);
  finalize_kernel<<<1, 256, 0, stream>>>(rs_aa, rs_bb, rs_ab, cs_ab, ta, out);
}